// MultivectorLoss_77232101916910
// MI455X (gfx1250) — compile-verified
//
#include <hip/hip_runtime.h>
#include <math.h>

typedef float v2f __attribute__((ext_vector_type(2)));
typedef float v4f __attribute__((ext_vector_type(4)));
typedef float v8f __attribute__((ext_vector_type(8)));
typedef int   i4v __attribute__((vector_size(16)));

#define B_         64
#define N_         16
#define LQ         32
#define LD         512
#define DIM        128
#define NDOC       17
#define CSTR       132               // padded LDS row stride (dwords): conflict-free b64 reads
#define CHUNK_ROWS 64
#define NCHUNK     (LD / CHUNK_ROWS) // 8
#define CHUNK_DW   (CHUNK_ROWS * CSTR)

#if __has_builtin(__builtin_amdgcn_global_load_async_to_lds_b128)
#define HAVE_ASYNC_LDS 1
#else
#define HAVE_ASYNC_LDS 0
#endif

// builtin signature (from compiler diagnostic): global int4* src, LDS int4* dst
typedef __attribute__((address_space(1))) i4v glb_i4;
typedef __attribute__((address_space(3))) i4v lds_i4;

__device__ __forceinline__ void copy_b128_to_lds(const float* gsrc, float* ldst)
{
#if HAVE_ASYNC_LDS
    // GLOBAL_LOAD_ASYNC_TO_LDS_B128: per-lane 16B global -> LDS, ASYNCcnt-tracked
    __builtin_amdgcn_global_load_async_to_lds_b128(
        (glb_i4*)gsrc, (lds_i4*)ldst, 0, 0);
#else
    *(v4f*)ldst = *(const v4f*)gsrc;   // fallback: synchronous VGPR round-trip
#endif
}

__device__ __forceinline__ void wait_async_copies()
{
#if HAVE_ASYNC_LDS
#if __has_builtin(__builtin_amdgcn_s_wait_asynccnt)
    __builtin_amdgcn_s_wait_asynccnt(0);
#else
    asm volatile("s_wait_asynccnt 0x0" ::: "memory");
#endif
#endif
}

__device__ __forceinline__ void stage_chunk(const float* dptr, int c, float* buf)
{
    // one 64x128 f32 chunk = contiguous 32KB; 256 threads x 8 b128 each,
    // written to LDS with padded row stride CSTR
    const float* g = dptr + (size_t)c * CHUNK_ROWS * DIM;
    const int t = threadIdx.x;
    #pragma unroll
    for (int j = 0; j < 8; ++j) {
        int seg = t + 256 * j;           // 2048 float4 segments
        int row = seg >> 5;              // 32 segs per 128-dword row
        int sc  = seg & 31;
        copy_b128_to_lds(g + seg * 4, buf + row * CSTR + sc * 4);
    }
}

__launch_bounds__(256, 2)
__global__ void maxsim_kernel(const float* __restrict__ q,
                              const float* __restrict__ pos,
                              const unsigned char* __restrict__ pmask,
                              const float* __restrict__ neg,
                              const unsigned char* __restrict__ nmask,
                              float* __restrict__ scores)
{
    __shared__ float dch[2 * CHUNK_DW];   // double-buffered doc chunk (2 x 33KB)
    __shared__ float rowmax_s[LQ];

    const int tid = threadIdx.x;
    const int b   = blockIdx.x / NDOC;
    const int doc = blockIdx.x % NDOC;

    const float* dptr;
    const unsigned char* mptr;
    if (doc == 0) {
        dptr = pos + (size_t)b * LD * DIM;
        mptr = pmask + (size_t)b * LD;
    } else {
        size_t off = (size_t)b * N_ + (doc - 1);
        dptr = neg + off * (size_t)(LD * DIM);
        mptr = nmask + off * LD;
    }

    // wave -> tile mapping: 8 waves = 2 M-tiles x 4 N-subtiles of the 64-col chunk
    const int w     = tid >> 5;
    const int lane  = tid & 31;
    const int mtile = w & 1;
    const int nloc  = w >> 1;
    const int hf    = lane >> 4;     // lane half selects K pair (ISA A/B layout)
    const int l16   = lane & 15;

    // kick off async fill of chunk 0 while we register-cache the A operand
    stage_chunk(dptr, 0, dch);

    // A operand (Q tile) cached in registers: invariant across all 8 chunks.
    // lane layout per ISA: row = mtile*16 + l16, K pair = {4s+2hf, 4s+2hf+1}
    const float* qrow = q + ((size_t)b * LQ + mtile * 16 + l16) * DIM + 2 * hf;
    v2f aq[32];
    #pragma unroll
    for (int s = 0; s < 32; ++s) aq[s] = *(const v2f*)&qrow[4 * s];

    if (tid < LQ) rowmax_s[tid] = -INFINITY;

    const int b_base = (nloc * 16 + l16) * CSTR + 2 * hf;

    float rmax[8];
    #pragma unroll
    for (int r = 0; r < 8; ++r) rmax[r] = -INFINITY;

    wait_async_copies();
    __syncthreads();                  // chunk 0 resident, rowmax_s initialized

    for (int c = 0; c < NCHUNK; ++c) {
        // overlap: async-fill the other buffer while computing this one
        if (c + 1 < NCHUNK)
            stage_chunk(dptr, c + 1, dch + ((c + 1) & 1) * CHUNK_DW);

        const float* bufc = dch + (c & 1) * CHUNK_DW;

        // 16x16 sim tile over K=128: 32x v_wmma_f32_16x16x4_f32,
        // one ds b64 (B operand) per WMMA, A from registers
        v8f acc = {};
        #pragma unroll
        for (int s = 0; s < 32; ++s) {
            v2f bv = *(const v2f*)&bufc[b_base + 4 * s];
            acc = __builtin_amdgcn_wmma_f32_16x16x4_f32(
                false, aq[s], false, bv, (short)0, acc, false, false);
        }

        // masked running max; this lane's column = doc token index
        int token = c * CHUNK_ROWS + nloc * 16 + l16;
        if (mptr[token]) {
            #pragma unroll
            for (int r = 0; r < 8; ++r) rmax[r] = fmaxf(rmax[r], acc[r]);
        }

        wait_async_copies();          // next chunk's fill complete (this wave)
        __syncthreads();              // all waves done computing + filling
    }

    // max over the 16 column-lanes of each half (xor stays within the half)
    #pragma unroll
    for (int off = 8; off >= 1; off >>= 1) {
        #pragma unroll
        for (int r = 0; r < 8; ++r)
            rmax[r] = fmaxf(rmax[r], __shfl_xor(rmax[r], off, 32));
    }
    // combine across the 4 waves that share this M-tile (disjoint column sets)
    if (l16 == 0) {
        #pragma unroll
        for (int r = 0; r < 8; ++r)
            atomicMax(&rowmax_s[mtile * 16 + hf * 8 + r], rmax[r]);
    }
    __syncthreads();

    // sum of per-query-row maxima -> score for this (b, doc)
    if (tid < 32) {
        float v = rowmax_s[tid];
        #pragma unroll
        for (int off = 16; off >= 1; off >>= 1) v += __shfl_xor(v, off, 32);
        if (tid == 0) scores[blockIdx.x] = v;   // blockIdx.x == b*NDOC + doc
    }
}

__global__ void loss_kernel(const float* __restrict__ scores,
                            float* __restrict__ out)
{
    __shared__ float partial[2];
    const int t = threadIdx.x;             // 0..63, one thread per batch row
    const float* s = scores + t * NDOC;

    float m = -INFINITY;
    #pragma unroll
    for (int i = 0; i < NDOC; ++i) m = fmaxf(m, s[i]);
    float sum = 0.0f;
    #pragma unroll
    for (int i = 0; i < NDOC; ++i) sum += expf(s[i] - m);
    float val = (m + logf(sum)) - s[0];    // logsumexp - positive score

    #pragma unroll
    for (int off = 16; off >= 1; off >>= 1) val += __shfl_xor(val, off, 32);
    if ((t & 31) == 0) partial[t >> 5] = val;
    __syncthreads();
    if (t == 0) out[0] = (partial[0] + partial[1]) * (1.0f / (float)B_);
}

extern "C" void kernel_launch(void* const* d_in, const int* in_sizes, int n_in,
                              void* d_out, int out_size, void* d_ws, size_t ws_size,
                              hipStream_t stream)
{
    const float*         q   = (const float*)d_in[0];
    const float*         pos = (const float*)d_in[1];
    const unsigned char* pm  = (const unsigned char*)d_in[2];
    const float*         neg = (const float*)d_in[3];
    const unsigned char* nm  = (const unsigned char*)d_in[4];

    float* scores = (float*)d_ws;          // B_*NDOC floats = 4352 B

    maxsim_kernel<<<dim3(B_ * NDOC), dim3(256), 0, stream>>>(q, pos, pm, neg, nm, scores);
    loss_kernel<<<dim3(1), dim3(64), 0, stream>>>(scores, (float*)d_out);
}